// MappingGNN_8572754723293
// MI455X (gfx1250) — compile-verified
//
#include <hip/hip_runtime.h>

#define N_NODES 100000
#define N_EDGES 6400000

typedef float v2f __attribute__((ext_vector_type(2)));
typedef float v8f __attribute__((ext_vector_type(8)));

// D(16x16,f32) += A(16x4,f32) x B(4x16,f32)  -- exact fp32 WMMA on gfx1250
__device__ __forceinline__ v8f wmma4(v2f a, v2f b, v8f c) {
  return __builtin_amdgcn_wmma_f32_16x16x4_f32(false, a, false, b, (short)0, c,
                                               false, false);
}

// ---------------- degree / normalization ----------------
__global__ __launch_bounds__(256) void k_degree(const int* __restrict__ dst,
                                                float* __restrict__ degf, int ne) {
  int e = blockIdx.x * 256 + threadIdx.x;
  if (e < ne) atomicAdd(&degf[dst[e]], 1.0f);
}

__global__ __launch_bounds__(256) void k_norm(const float* __restrict__ degf,
                                              float* __restrict__ dinv,
                                              float* __restrict__ rmean, int n) {
  int i = blockIdx.x * 256 + threadIdx.x;
  if (i < n) {
    float d = degf[i];
    dinv[i]  = rsqrtf(d + 1.0f);        // GCN deg includes the self loop, always > 0
    rmean[i] = 1.0f / fmaxf(d, 1.0f);   // SAGE mean divisor
  }
}

// ---------------- dense 16x16 GEMM via f32 WMMA: Y = X @ W ----------------
__global__ __launch_bounds__(256) void k_gemm16(const float* __restrict__ X,
                                                const float* __restrict__ W,
                                                float* __restrict__ Y, int n) {
  int lane = threadIdx.x & 31;
  int wv   = threadIdx.x >> 5;
  int ntiles = n >> 4;
  int tile = blockIdx.x * 8 + wv;
  if (tile >= ntiles) tile = ntiles - 1;   // clamp: keep EXEC all-ones for WMMA
  int base = tile << 4;
  int half = lane >> 4, l16 = lane & 15, koff = half << 1;
  v8f acc = {};
#pragma unroll
  for (int kc = 0; kc < 4; ++kc) {
    int k0 = kc * 4 + koff;
    v2f a, b;
    a.x = X[(base + l16) * 16 + k0];
    a.y = X[(base + l16) * 16 + k0 + 1];
    b.x = W[k0 * 16 + l16];
    b.y = W[(k0 + 1) * 16 + l16];
    acc = wmma4(a, b, acc);
  }
#pragma unroll
  for (int r = 0; r < 8; ++r)
    Y[(base + r + 8 * half) * 16 + l16] = acc[r];
}

// ---------------- edge scatters (4 lanes per edge, float4 gather) ----------------
__global__ __launch_bounds__(256) void k_gcn_scatter(const int* __restrict__ src,
                                                     const int* __restrict__ dst,
                                                     const float* __restrict__ xw,
                                                     const float* __restrict__ dinv,
                                                     float* __restrict__ acc, int ne) {
  long long t = (long long)blockIdx.x * 256 + threadIdx.x;
  int e = (int)(t >> 2), q = (int)(t & 3);
  if (e >= ne) return;
  int s = src[e], d = dst[e];
  float nrm = dinv[s] * dinv[d];
  float4 v = reinterpret_cast<const float4*>(xw)[s * 4 + q];
  float* a = acc + d * 16 + q * 4;
  atomicAdd(a + 0, v.x * nrm);
  atomicAdd(a + 1, v.y * nrm);
  atomicAdd(a + 2, v.z * nrm);
  atomicAdd(a + 3, v.w * nrm);
}

__global__ __launch_bounds__(256) void k_sage_scatter(const int* __restrict__ src,
                                                      const int* __restrict__ dst,
                                                      const float* __restrict__ h,
                                                      float* __restrict__ agg, int ne) {
  long long t = (long long)blockIdx.x * 256 + threadIdx.x;
  int e = (int)(t >> 2), q = (int)(t & 3);
  if (e >= ne) return;
  int s = src[e], d = dst[e];
  float4 v = reinterpret_cast<const float4*>(h)[s * 4 + q];
  float* a = agg + d * 16 + q * 4;
  atomicAdd(a + 0, v.x);
  atomicAdd(a + 1, v.y);
  atomicAdd(a + 2, v.z);
  atomicAdd(a + 3, v.w);
}

// h1 = tanh(acc + self-loop term + bias), in place
__global__ __launch_bounds__(256) void k_gcn_fin(float* __restrict__ acc,
                                                 const float* __restrict__ xw,
                                                 const float* __restrict__ dinv,
                                                 const float* __restrict__ b, int n) {
  int i = blockIdx.x * 256 + threadIdx.x;
  if (i < n * 16) {
    int node = i >> 4, f = i & 15;
    float di = dinv[node];
    acc[i] = tanhf(acc[i] + xw[i] * di * di + b[f]);
  }
}

// ---------------- SAGE: out = L2norm((agg*rmean)@Wl + bl + h@Wr), opt tanh ----------------
__global__ __launch_bounds__(256) void k_sage(const float* __restrict__ h,
                                              const float* __restrict__ agg,
                                              const float* __restrict__ rmean,
                                              const float* __restrict__ Wl,
                                              const float* __restrict__ bl,
                                              const float* __restrict__ Wr,
                                              float* __restrict__ out, int n, int do_tanh) {
  __shared__ float lds[8][16][17];
  int lane = threadIdx.x & 31;
  int wv   = threadIdx.x >> 5;
  int ntiles = n >> 4;
  int tile = blockIdx.x * 8 + wv;
  if (tile >= ntiles) tile = ntiles - 1;
  int base = tile << 4;
  int half = lane >> 4, l16 = lane & 15, koff = half << 1;
  int node = base + l16;
  float rm = rmean[node];
  v8f acc = {};
#pragma unroll
  for (int kc = 0; kc < 4; ++kc) {         // (agg * rmean) @ Wl
    int k0 = kc * 4 + koff;
    v2f a, b;
    a.x = agg[node * 16 + k0] * rm;
    a.y = agg[node * 16 + k0 + 1] * rm;
    b.x = Wl[k0 * 16 + l16];
    b.y = Wl[(k0 + 1) * 16 + l16];
    acc = wmma4(a, b, acc);
  }
#pragma unroll
  for (int kc = 0; kc < 4; ++kc) {         // + h @ Wr
    int k0 = kc * 4 + koff;
    v2f a, b;
    a.x = h[node * 16 + k0];
    a.y = h[node * 16 + k0 + 1];
    b.x = Wr[k0 * 16 + l16];
    b.y = Wr[(k0 + 1) * 16 + l16];
    acc = wmma4(a, b, acc);
  }
  float bias = bl[l16];
#pragma unroll
  for (int r = 0; r < 8; ++r)
    lds[wv][r + 8 * half][l16] = acc[r] + bias;
  asm volatile("s_wait_dscnt 0" ::: "memory");
  if (lane < 16) {                         // per-row L2 normalize
    float ss = 0.f;
#pragma unroll
    for (int f = 0; f < 16; ++f) { float v = lds[wv][lane][f]; ss += v * v; }
    float inv = 1.0f / fmaxf(sqrtf(ss), 1e-12f);
#pragma unroll
    for (int f = 0; f < 16; ++f) {
      float v = lds[wv][lane][f] * inv;
      out[(base + lane) * 16 + f] = do_tanh ? tanhf(v) : v;
    }
  }
}

// ---------------- fused MLP: relu(h@W1+b1) -> relu(@W2+b2) -> @W3+b3 ----------------
__global__ __launch_bounds__(128) void k_mlp(const float* __restrict__ h,
                                             const float* __restrict__ W1,
                                             const float* __restrict__ b1,
                                             const float* __restrict__ W2,
                                             const float* __restrict__ b2,
                                             const float* __restrict__ W3,
                                             const float* __restrict__ b3,
                                             float* __restrict__ out, int n) {
  __shared__ float y1[4][16][128];
  __shared__ float part[4][16][16];
  int lane = threadIdx.x & 31;
  int wv   = threadIdx.x >> 5;
  int ntiles = n >> 4;
  int tile = blockIdx.x * 4 + wv;
  if (tile >= ntiles) tile = ntiles - 1;
  int base = tile << 4;
  int half = lane >> 4, l16 = lane & 15, koff = half << 1;

  // layer 1: y1 = relu(h@W1 + b1)   (16x16 @ 16x128)
#pragma unroll
  for (int j = 0; j < 8; ++j) {
    v8f acc = {};
#pragma unroll
    for (int kc = 0; kc < 4; ++kc) {
      int k0 = kc * 4 + koff;
      v2f a, b;
      a.x = h[(base + l16) * 16 + k0];
      a.y = h[(base + l16) * 16 + k0 + 1];
      b.x = W1[k0 * 128 + j * 16 + l16];
      b.y = W1[(k0 + 1) * 128 + j * 16 + l16];
      acc = wmma4(a, b, acc);
    }
    float bias = b1[j * 16 + l16];
#pragma unroll
    for (int r = 0; r < 8; ++r)
      y1[wv][r + 8 * half][j * 16 + l16] = fmaxf(acc[r] + bias, 0.0f);
  }
  asm volatile("s_wait_dscnt 0" ::: "memory");

  // layer 2 (16x128 @ 128x128) with layer-3 dot folded into partials
  float p[8];
#pragma unroll
  for (int r = 0; r < 8; ++r) p[r] = 0.f;
  for (int j = 0; j < 8; ++j) {
    v8f acc = {};
    for (int kc = 0; kc < 32; ++kc) {
      int k0 = kc * 4 + koff;
      v2f a, b;
      a.x = y1[wv][l16][k0];
      a.y = y1[wv][l16][k0 + 1];
      b.x = W2[k0 * 128 + j * 16 + l16];
      b.y = W2[(k0 + 1) * 128 + j * 16 + l16];
      acc = wmma4(a, b, acc);
    }
    float bias = b2[j * 16 + l16];
    float w3c  = W3[j * 16 + l16];
#pragma unroll
    for (int r = 0; r < 8; ++r)
      p[r] += fmaxf(acc[r] + bias, 0.0f) * w3c;
  }
#pragma unroll
  for (int r = 0; r < 8; ++r) part[wv][r + 8 * half][l16] = p[r];
  asm volatile("s_wait_dscnt 0" ::: "memory");
  if (lane < 16) {
    float s = 0.f;
#pragma unroll
    for (int c = 0; c < 16; ++c) s += part[wv][lane][c];
    out[base + lane] = s + b3[0];
  }
}

extern "C" void kernel_launch(void* const* d_in, const int* in_sizes, int n_in,
                              void* d_out, int out_size, void* d_ws, size_t ws_size,
                              hipStream_t stream) {
  (void)in_sizes; (void)n_in; (void)out_size; (void)ws_size;
  const float* x   = (const float*)d_in[0];
  const int*   ei  = (const int*)d_in[1];     // JAX x64 disabled -> int32 indices
  const float* Wg  = (const float*)d_in[2];
  const float* bg  = (const float*)d_in[3];
  const float* Wl1 = (const float*)d_in[4];
  const float* bl1 = (const float*)d_in[5];
  const float* Wr1 = (const float*)d_in[6];
  const float* Wl2 = (const float*)d_in[7];
  const float* bl2 = (const float*)d_in[8];
  const float* Wr2 = (const float*)d_in[9];
  const float* W1  = (const float*)d_in[10];
  const float* b1  = (const float*)d_in[11];
  const float* W2  = (const float*)d_in[12];
  const float* b2  = (const float*)d_in[13];
  const float* W3  = (const float*)d_in[14];
  const float* b3  = (const float*)d_in[15];
  float* out = (float*)d_out;

  const int n = N_NODES, ne = N_EDGES;
  const int* src = ei;
  const int* dst = ei + ne;

  float* ws    = (float*)d_ws;
  float* degf  = ws;
  float* dinv  = ws + n;
  float* rmean = ws + 2 * n;
  float* bufA  = ws + 3 * n;          // xw / h2
  float* bufB  = bufA + n * 16;       // gcn acc -> h1 -> h3
  float* bufC  = bufB + n * 16;       // sage agg

  int ntiles = n / 16;
  int eblk   = (int)(((long long)ne * 4 + 255) / 256);

  // degrees + norms
  hipMemsetAsync(degf, 0, (size_t)n * sizeof(float), stream);
  k_degree<<<(ne + 255) / 256, 256, 0, stream>>>(dst, degf, ne);
  k_norm<<<(n + 255) / 256, 256, 0, stream>>>(degf, dinv, rmean, n);

  // GCN layer
  k_gemm16<<<(ntiles + 7) / 8, 256, 0, stream>>>(x, Wg, bufA, n);
  hipMemsetAsync(bufB, 0, (size_t)n * 16 * sizeof(float), stream);
  k_gcn_scatter<<<eblk, 256, 0, stream>>>(src, dst, bufA, dinv, bufB, ne);
  k_gcn_fin<<<(n * 16 + 255) / 256, 256, 0, stream>>>(bufB, bufA, dinv, bg, n);

  // SAGE layer 1 (tanh)
  hipMemsetAsync(bufC, 0, (size_t)n * 16 * sizeof(float), stream);
  k_sage_scatter<<<eblk, 256, 0, stream>>>(src, dst, bufB, bufC, ne);
  k_sage<<<(ntiles + 7) / 8, 256, 0, stream>>>(bufB, bufC, rmean, Wl1, bl1, Wr1, bufA, n, 1);

  // SAGE layer 2 (no tanh)
  hipMemsetAsync(bufC, 0, (size_t)n * 16 * sizeof(float), stream);
  k_sage_scatter<<<eblk, 256, 0, stream>>>(src, dst, bufA, bufC, ne);
  k_sage<<<(ntiles + 7) / 8, 256, 0, stream>>>(bufA, bufC, rmean, Wl2, bl2, Wr2, bufB, n, 0);

  // fused MLP head
  k_mlp<<<(ntiles + 3) / 4, 128, 0, stream>>>(bufB, W1, b1, W2, b2, W3, b3, out, n);
}